// BiGRIL_37606733643899
// MI455X (gfx1250) — compile-verified
//
#include <hip/hip_runtime.h>

// BiGRIL reduced: h==0 collapses everything to
//   Xg = adj^T @ x1, Mg = adj^T @ m  (per batch), colsum = 1^T adj
//   o_pre[p] = d0[p]*x1 + d1[p]*m + d2[p]*Xg + d3[p]*Mg + d4[p]*colsum + d5[p]
//   out = W_o2 . relu(W_o1 * (W_ro . prelu(o_pre) + b_ro) + b_o1) + b_o2
// GEMM done with V_WMMA_F32_16X16X4_F32 (exact f32, chip is HBM-bound here).

#define B_   8
#define N_   1024
#define T_   64
#define H_   64
#define NT_  (N_ * T_)        // 65536
#define TOT_ (B_ * NT_)       // 524288

typedef __attribute__((ext_vector_type(2))) float v2f;
typedef __attribute__((ext_vector_type(8))) float v8f;

// workspace layout (float offsets)
#define WS_RHS  0u                         // [2][8][1024][64]  (g=0: x1, g=1: mask)
#define WS_GOUT (WS_RHS + 2u * TOT_)       // [2][8][1024][64]  (g=0: Xg, g=1: Mg)
#define WS_COEF (WS_GOUT + 2u * TOT_)      // d0..d5 : 6*64
#define WS_COL  (WS_COEF + 6u * 64u)       // colsum : 1024

// ---------------------------------------------------------------- prep RHS
__global__ void bigril_prep_rhs(const float* __restrict__ x,
                                const int* __restrict__ mask,
                                const float* __restrict__ b_fs,
                                float* __restrict__ ws) {
    int idx = blockIdx.x * 256 + threadIdx.x;   // < TOT_
    float bfs = b_fs[0];
    int   m   = mask[idx];
    float xv  = x[idx];
    ws[WS_RHS + idx]        = m ? xv : bfs;     // x1
    ws[WS_RHS + TOT_ + idx] = (float)m;         // m_f
}

// ---------------------------------------------------------------- colsum
__global__ void bigril_colsum(const float* __restrict__ adj,
                              float* __restrict__ ws) {
    int m = blockIdx.x * 256 + threadIdx.x;     // < 1024
    float s = 0.f;
    for (int n = 0; n < N_; ++n) s += adj[n * N_ + m];
    ws[WS_COL + m] = s;
}

// ---------------------------------------------------------------- coeff fold
// e* = W_gc halves applied to {w0, w1, b_in}; d* = W_lo[:, :H] @ e* (+bias)
__global__ void bigril_coeffs(const float* __restrict__ W_in,
                              const float* __restrict__ b_in,
                              const float* __restrict__ W_gc,
                              const float* __restrict__ b_gc,
                              const float* __restrict__ W_lo,
                              const float* __restrict__ b_lo,
                              float* __restrict__ ws) {
    __shared__ float w0s[64], w1s[64], bis[64];
    __shared__ float e[6][64];
    int o = threadIdx.x;                         // 64 threads
    w0s[o] = W_in[o * 66 + 0];
    w1s[o] = W_in[o * 66 + 1];
    bis[o] = b_in[o];
    __syncthreads();
    float e0 = 0.f, e1 = 0.f, e2 = 0.f, e3 = 0.f, e4 = 0.f, e5 = 0.f;
    for (int c = 0; c < 64; ++c) {
        float g1 = W_gc[o * 128 + c];
        float g2 = W_gc[o * 128 + 64 + c];
        e0 += g1 * w0s[c];  e1 += g1 * w1s[c];
        e2 += g2 * w0s[c];  e3 += g2 * w1s[c];
        e4 += g2 * bis[c];  e5 += g1 * bis[c];
    }
    e[0][o] = e0; e[1][o] = e1; e[2][o] = e2;
    e[3][o] = e3; e[4][o] = e4; e[5][o] = e5 + b_gc[o];
    __syncthreads();
    int p = o;
    float d0 = 0.f, d1 = 0.f, d2 = 0.f, d3 = 0.f, d4 = 0.f, d5 = 0.f;
    for (int c = 0; c < 64; ++c) {
        float l = W_lo[p * 128 + c];
        d0 += l * e[0][c]; d1 += l * e[1][c]; d2 += l * e[2][c];
        d3 += l * e[3][c]; d4 += l * e[4][c]; d5 += l * e[5][c];
    }
    ws[WS_COEF + 0 * 64 + p] = d0;
    ws[WS_COEF + 1 * 64 + p] = d1;
    ws[WS_COEF + 2 * 64 + p] = d2;
    ws[WS_COEF + 3 * 64 + p] = d3;
    ws[WS_COEF + 4 * 64 + p] = d4;
    ws[WS_COEF + 5 * 64 + p] = d5 + b_lo[p];
}

// ---------------------------------------------------------------- WMMA GEMM
// gout[y][m][t] = sum_n adj[n][m] * rhs[y][n][t]; y = g*8+b in [0,16).
// One wave per (m-tile, y); 4 accumulators cover t = 0..63.
__global__ __launch_bounds__(32) void bigril_gemm(const float* __restrict__ adj,
                                                  float* __restrict__ ws) {
    const float* rhs  = ws + WS_RHS  + (size_t)blockIdx.y * NT_;
    float*       gout = ws + WS_GOUT + (size_t)blockIdx.y * NT_;
    const int m0   = blockIdx.x * 16;
    const int lane = threadIdx.x;
    const int lm   = lane & 15;            // M index (A), N/t index (B, C/D)
    const int kk   = (lane >> 4) << 1;     // K base for this half-wave

    v8f acc0 = {}, acc1 = {}, acc2 = {}, acc3 = {};

    for (int k0 = 0; k0 < N_; k0 += 4) {
        const int ka = k0 + kk;
        v2f a;                               // A = adj^T tile: A[m,k] = adj[k][m]
        a.x = adj[(size_t)ka * N_ + m0 + lm];
        a.y = adj[(size_t)(ka + 1) * N_ + m0 + lm];
        const float* r0 = rhs + (size_t)ka * T_ + lm;
        const float* r1 = rhs + (size_t)(ka + 1) * T_ + lm;
        v2f b0; b0.x = r0[0];  b0.y = r1[0];
        v2f b1; b1.x = r0[16]; b1.y = r1[16];
        v2f b2; b2.x = r0[32]; b2.y = r1[32];
        v2f b3; b3.x = r0[48]; b3.y = r1[48];
        acc0 = __builtin_amdgcn_wmma_f32_16x16x4_f32(false, a, false, b0, (short)0, acc0, false, false);
        acc1 = __builtin_amdgcn_wmma_f32_16x16x4_f32(false, a, false, b1, (short)0, acc1, false, false);
        acc2 = __builtin_amdgcn_wmma_f32_16x16x4_f32(false, a, false, b2, (short)0, acc2, false, false);
        acc3 = __builtin_amdgcn_wmma_f32_16x16x4_f32(false, a, false, b3, (short)0, acc3, false, false);
    }

    const int hi = (lane >> 4) * 8;          // C/D layout: lanes 16-31 hold M+8
#pragma unroll
    for (int r = 0; r < 8; ++r) {
        const int row = m0 + r + hi;
        gout[(size_t)row * T_ + 0  + lm] = acc0[r];
        gout[(size_t)row * T_ + 16 + lm] = acc1[r];
        gout[(size_t)row * T_ + 32 + lm] = acc2[r];
        gout[(size_t)row * T_ + 48 + lm] = acc3[r];
    }
}

// ---------------------------------------------------------------- epilogue
__global__ void bigril_final(const float* __restrict__ ws,
                             float* __restrict__ out,
                             const float* __restrict__ W_ro,
                             const float* __restrict__ b_ro,
                             const float* __restrict__ prelu_a,
                             const float* __restrict__ W_o1,
                             const float* __restrict__ b_o1,
                             const float* __restrict__ W_o2,
                             const float* __restrict__ b_o2) {
    __shared__ float sd[6][64], sro[64], so1[64], sb1[64], so2[64];
    const int tid = threadIdx.x;
    if (tid < 64) {
#pragma unroll
        for (int j = 0; j < 6; ++j) sd[j][tid] = ws[WS_COEF + j * 64 + tid];
        sro[tid] = W_ro[tid];          // only first H cols matter (rest hit h==0)
        so1[tid] = W_o1[tid];          // (FF,1)
        sb1[tid] = b_o1[tid];
        so2[tid] = W_o2[tid];          // (1,FF)
    }
    __syncthreads();

    const int idx = blockIdx.x * 256 + tid;      // < TOT_
    const int n   = (idx >> 6) & (N_ - 1);

    const float x1v = ws[WS_RHS + idx];
    const float mv  = ws[WS_RHS + TOT_ + idx];
    const float Xg  = ws[WS_GOUT + idx];
    const float Mg  = ws[WS_GOUT + TOT_ + idx];
    const float cs  = ws[WS_COL + n];
    const float a   = prelu_a[0];

    float acc = b_ro[0];
#pragma unroll 8
    for (int p = 0; p < 64; ++p) {
        float v = sd[0][p] * x1v + sd[1][p] * mv + sd[2][p] * Xg +
                  sd[3][p] * Mg  + sd[4][p] * cs + sd[5][p];
        v = (v >= 0.f) ? v : a * v;              // PReLU
        acc += sro[p] * v;
    }

    float res = b_o2[0];
#pragma unroll 8
    for (int f = 0; f < 64; ++f) {
        float y = so1[f] * acc + sb1[f];
        y = (y > 0.f) ? y : 0.f;                 // ReLU
        res += so2[f] * y;
    }
    out[idx] = res;
}

// ---------------------------------------------------------------- launch
extern "C" void kernel_launch(void* const* d_in, const int* in_sizes, int n_in,
                              void* d_out, int out_size, void* d_ws, size_t ws_size,
                              hipStream_t stream) {
    const float* x    = (const float*)d_in[0];
    const int*   mask = (const int*)d_in[1];
    // d_in[2] = W_fs (multiplies h==0, unused)
    const float* b_fs = (const float*)d_in[3];
    const float* W_in = (const float*)d_in[4];
    const float* b_in = (const float*)d_in[5];
    const float* adj  = (const float*)d_in[6];
    const float* W_gc = (const float*)d_in[7];
    const float* b_gc = (const float*)d_in[8];
    const float* W_lo = (const float*)d_in[9];
    const float* b_lo = (const float*)d_in[10];
    const float* pa   = (const float*)d_in[11];
    const float* W_ro = (const float*)d_in[12];
    const float* b_ro = (const float*)d_in[13];
    const float* W_o1 = (const float*)d_in[14];
    const float* b_o1 = (const float*)d_in[15];
    const float* W_o2 = (const float*)d_in[16];
    const float* b_o2 = (const float*)d_in[17];
    float* out = (float*)d_out;
    float* ws  = (float*)d_ws;

    bigril_prep_rhs<<<TOT_ / 256, 256, 0, stream>>>(x, mask, b_fs, ws);
    bigril_colsum  <<<N_ / 256,   256, 0, stream>>>(adj, ws);
    bigril_coeffs  <<<1,           64, 0, stream>>>(W_in, b_in, W_gc, b_gc, W_lo, b_lo, ws);
    bigril_gemm    <<<dim3(N_ / 16, 16), 32, 0, stream>>>(adj, ws);
    bigril_final   <<<TOT_ / 256, 256, 0, stream>>>(ws, out, W_ro, b_ro, pa,
                                                    W_o1, b_o1, W_o2, b_o2);
}